// DPLayer_89773406421536
// MI455X (gfx1250) — compile-verified
//
#include <hip/hip_runtime.h>
#include <stdint.h>

// Shortest-path DP on a 64x64 grid DAG, one wave32 per sample.
// Min-plus row scan done with an exact Kogge-Stone over lane pairs.
// Rows are staged global->LDS with gfx1250 async copies, double-buffered.

#define BIG 1e9f

__device__ __forceinline__ void async_row_load(uint32_t ldsAddr, const float* gptr) {
    // Each lane copies 8 bytes (2 floats) of the row into LDS asynchronously.
    // LDS destination is given by the ldsAddr VGPR; tracked with ASYNCcnt.
    uint64_t g = (uint64_t)(uintptr_t)gptr;
    asm volatile("global_load_async_to_lds_b64 %0, %1, off"
                 :: "v"(ldsAddr), "v"(g)
                 : "memory");
}

__global__ __launch_bounds__(256) void dp_sp_kernel(const float* __restrict__ images,
                                                    float* __restrict__ out, int B) {
    constexpr int H = 64, W = 64;
    const int lane = threadIdx.x & 31;
    const int wv   = threadIdx.x >> 5;
    const int s    = blockIdx.x * 8 + wv;
    if (s >= B) return;

    __shared__ float smem[8][2][W];   // 8 waves x double buffer x one row (4 KB)

    const float* img = images + (size_t)s * (H * W);
    const int j0 = 2 * lane;          // this lane owns columns j0, j0+1

    // Low 32 bits of the flat shared address == LDS byte offset (aperture layout).
    uint32_t ldsA[2];
    ldsA[0] = (uint32_t)(uintptr_t)&smem[wv][0][j0];
    ldsA[1] = (uint32_t)(uintptr_t)&smem[wv][1][j0];

    async_row_load(ldsA[0], img + j0);            // prime row 0

    float pd0 = 0.f, pd1 = 0.f;                   // prev-row DP values at j0, j0+1
    float pi0 = 0.f, pi1 = 0.f;                   // prev-row image values

    for (int i = 0; i < H; ++i) {
        const int cur = i & 1;
        if (i + 1 < H) {
            async_row_load(ldsA[cur ^ 1], img + (i + 1) * W + j0);
            // async loads complete in order: <=1 outstanding => row i is in LDS
            asm volatile("s_wait_asynccnt 0x1" ::: "memory");
        } else {
            asm volatile("s_wait_asynccnt 0x0" ::: "memory");
        }

        float2 cv = *(const float2*)&smem[wv][cur][j0];
        const float ci0 = cv.x, ci1 = cv.y;       // current-row image at j0, j0+1

        const float ci1m = __shfl_up(ci1, 1);     // cur img at j0-1 (from lane-1)

        float c0, c1;                             // candidates before right-edge scan
        if (i == 0) {
            c0 = (lane == 0) ? 0.0f : BIG;        // only (0,0) is a source
            c1 = BIG;
        } else {
            const float pi1m = __shfl_up(pi1, 1); // prev img at j0-1
            const float pd1m = __shfl_up(pd1, 1); // prev d   at j0-1
            c0 = pd0 + 0.5f * (pi0 + ci0);                        // down edge
            c1 = pd1 + 0.5f * (pi1 + ci1);                        // down edge
            if (lane > 0) c0 = fminf(c0, pd1m + 0.5f * (pi1m + ci0)); // diag edge
            c1 = fminf(c1, pd0 + 0.5f * (pi0 + ci1));                 // diag edge
        }

        // Right-edge weights: a[j] = 0.5*(img[j-1]+img[j]); a[0] = BIG sentinel.
        const float a0 = (lane == 0) ? BIG : 0.5f * (ci1m + ci0);
        const float a1 = 0.5f * (ci0 + ci1);

        // Min-plus scan of pairs (a,c):
        //   combine((a1,c1),(a2,c2)) = (a1+a2, min(c2, c1+a2))
        // Serial combine of the lane's two elements:
        float A = a0 + a1;
        float C = fminf(c1, c0 + a1);
        // Exact Kogge-Stone inclusive scan across the 32 lanes (segments never
        // overlap under the lane>=d guard, so the non-idempotent '+' part is exact).
        #pragma unroll
        for (int d = 1; d < 32; d <<= 1) {
            const float Au = __shfl_up(A, d);
            const float Cu = __shfl_up(C, d);
            if (lane >= d) {
                C = fminf(C, Cu + A);   // uses pre-update A (right operand)
                A = Au + A;
            }
        }
        // Exclusive prefix (through column j0-1), then per-lane outputs.
        float Cex = __shfl_up(C, 1);
        if (lane == 0) Cex = BIG;
        const float v0 = fminf(c0, Cex + a0);
        const float v1 = fminf(c1, v0 + a1);

        pd0 = v0; pd1 = v1; pi0 = ci0; pi1 = ci1;
    }

    if (lane == 31) out[s] = pd1;     // d[H-1][W-1]
}

extern "C" void kernel_launch(void* const* d_in, const int* in_sizes, int n_in,
                              void* d_out, int out_size, void* d_ws, size_t ws_size,
                              hipStream_t stream) {
    const float* images = (const float*)d_in[0];
    float* out = (float*)d_out;
    const int B = in_sizes[0] / (64 * 64);      // 2048
    const int blocks = (B + 7) / 8;             // 8 waves (samples) per block
    dp_sp_kernel<<<blocks, 256, 0, stream>>>(images, out, B);
}